// ANet_sm_57191784513797
// MI455X (gfx1250) — compile-verified
//
#include <hip/hip_runtime.h>
#include <math.h>

typedef __attribute__((ext_vector_type(2))) float v2f;
typedef __attribute__((ext_vector_type(8))) float v8f;

// ---------------------------------------------------------------------------
// WMMA fp32 GEMM: C(MxN) = A(MxK) @ B(KxN) [+bias][relu], row-major.
// REQUIRES K % 4 == 0 (all call sites: K = 36, 512, 256).
// One wave per 16x16 C tile, K stepped by 4 (V_WMMA_F32_16X16X4_F32).
// Branch-free fragment loads: clamped addresses + loop-invariant selects so
// EXEC stays all-1s (WMMA requirement) and the inner loop is load+load+wmma.
// A frag (16x4): lanes0-15 K={k,k+1}, lanes16-31 K={k+2,k+3}.
// B frag (4x16): same K halves, N striped across lanes.
// C/D (16x16): vgpr g -> M = g (+8 for upper half), N = lane&15.
// ---------------------------------------------------------------------------
__global__ void k_gemm_wmma(const float* __restrict__ A, const float* __restrict__ B,
                            float* __restrict__ C, int M, int N, int K,
                            const float* __restrict__ bias, int relu) {
  int ntiles = (N + 15) >> 4;
  int mt = blockIdx.x / ntiles;
  int nt = blockIdx.x % ntiles;
  int lane = threadIdx.x;
  int half = lane >> 4;
  int l16  = lane & 15;
  int m = mt * 16 + l16;
  int n = nt * 16 + l16;
  bool mok = m < M;
  bool nok = n < N;
  int mc = mok ? m : (M - 1);
  int nc = nok ? n : (N - 1);
  // per-lane base pointers (include the K-half offset for this lane)
  const float* Arow = A + (size_t)mc * K + half * 2;       // reads [k0], [k0+1]
  const float* Bcol = B + (size_t)(half * 2) * N + nc;     // reads [k0*N], [k0*N+N]
  v8f acc = {};
  for (int k0 = 0; k0 < K; k0 += 4) {
    __builtin_prefetch(Bcol + (size_t)(k0 + 4) * N, 0, 1);
    v2f a = *(const v2f*)(Arow + k0);                       // one b64 load
    float b0 = Bcol[(size_t)k0 * N];
    float b1 = Bcol[(size_t)k0 * N + N];
    a.x = mok ? a.x : 0.f;
    a.y = mok ? a.y : 0.f;
    v2f b;
    b.x = nok ? b0 : 0.f;
    b.y = nok ? b1 : 0.f;
    acc = __builtin_amdgcn_wmma_f32_16x16x4_f32(false, a, false, b, (short)0, acc,
                                                false, false);
  }
#pragma unroll
  for (int g = 0; g < 8; ++g) {
    int mm = mt * 16 + g + half * 8;
    if (mm < M && n < N) {
      float v = acc[g];
      if (bias) v += bias[mm];
      if (relu) v = fmaxf(v, 0.f);
      C[(size_t)mm * N + n] = v;
    }
  }
}

// ---------------------------------------------------------------------------
// Stem helpers: Wv_eff[o][t] = sum_m softmax_m(emb)[m][t] * wv[m][o]
// ---------------------------------------------------------------------------
__global__ void k_wveff(const float* __restrict__ emb_a, const float* __restrict__ emb_b,
                        const float* __restrict__ emb_c, const float* __restrict__ emb_mix,
                        const float* __restrict__ s_wv, float* __restrict__ wveff) {
  int t = threadIdx.x;
  if (t >= 64) return;
  int a = t >> 4, b = (t >> 2) & 3, g = t & 3;
  float e[4];
  float mx = -INFINITY;
  for (int m = 0; m < 4; ++m) {
    float la = 0.f, lb = 0.f, lc = 0.f;
    for (int c = 0; c < 36; ++c) {
      float mv = emb_mix[m * 36 + c];
      la += mv * emb_a[c * 4 + a];
      lb += mv * emb_b[c * 4 + b];
      lc += mv * emb_c[c * 4 + g];
    }
    e[m] = la + lb + lc;
    mx = fmaxf(mx, e[m]);
  }
  float se = 0.f;
  for (int m = 0; m < 4; ++m) { e[m] = __expf(e[m] - mx); se += e[m]; }
  for (int o = 0; o < 36; ++o) {
    float acc = 0.f;
    for (int m = 0; m < 4; ++m) acc += e[m] * s_wv[m * 36 + o];
    wveff[o * 64 + t] = acc / se;
  }
}

// Stem attention: out[c, d,h,w] over 64 taps (k=4, pad=1) at 32^3.
// logit[t] = (wq[c]*x0*wk[c]) * x[tap t]  -> factor s common, softmax register-light.
__global__ void k_stem(const float* __restrict__ x, const float* __restrict__ wq,
                       const float* __restrict__ wk, const float* __restrict__ wveff,
                       float* __restrict__ out) {
  const int NP = 32 * 32 * 32;
  int idx = blockIdx.x * blockDim.x + threadIdx.x;
  if (idx >= 36 * NP) return;
  int c = idx / NP, p = idx % NP;
  int d = p >> 10, h = (p >> 5) & 31, w = p & 31;
  float x0 = x[(d * 33 + h) * 33 + w];
  float s = wq[c] * x0 * wk[c];
  float mn = INFINITY, mx = -INFINITY;
#pragma unroll
  for (int i = 0; i < 4; ++i)
#pragma unroll
    for (int j = 0; j < 4; ++j)
#pragma unroll
      for (int l = 0; l < 4; ++l) {
        int dd = d + i - 1, hh = h + j - 1, ww = w + l - 1;
        float v = 0.f;
        if (dd >= 0 && dd < 33 && hh >= 0 && hh < 33 && ww >= 0 && ww < 33)
          v = x[(dd * 33 + hh) * 33 + ww];
        mn = fminf(mn, v);
        mx = fmaxf(mx, v);
      }
  float mlog = fmaxf(s * mn, s * mx);
  float se = 0.f, num = 0.f;
#pragma unroll
  for (int i = 0; i < 4; ++i)
#pragma unroll
    for (int j = 0; j < 4; ++j)
#pragma unroll
      for (int l = 0; l < 4; ++l) {
        int dd = d + i - 1, hh = h + j - 1, ww = w + l - 1;
        float v = 0.f;
        if (dd >= 0 && dd < 33 && hh >= 0 && hh < 33 && ww >= 0 && ww < 33)
          v = x[(dd * 33 + hh) * 33 + ww];
        int t = (i * 4 + j) * 4 + l;
        float e = __expf(s * v - mlog);
        se += e;
        num += e * (wveff[c * 64 + t] * v);
      }
  out[idx] = num / se;
}

// ---------------------------------------------------------------------------
// BN stats (one block per channel) + fused BN/ReLU/maxpool2
// ---------------------------------------------------------------------------
__global__ void k_bnstats(const float* __restrict__ x, float* __restrict__ stats, int NP) {
  __shared__ float ss[256], sq[256];
  int c = blockIdx.x;
  float s = 0.f, q = 0.f;
  for (int p = threadIdx.x; p < NP; p += blockDim.x) {
    float v = x[c * NP + p];
    s += v; q += v * v;
  }
  ss[threadIdx.x] = s; sq[threadIdx.x] = q;
  __syncthreads();
  for (int off = blockDim.x >> 1; off > 0; off >>= 1) {
    if ((int)threadIdx.x < off) {
      ss[threadIdx.x] += ss[threadIdx.x + off];
      sq[threadIdx.x] += sq[threadIdx.x + off];
    }
    __syncthreads();
  }
  if (threadIdx.x == 0) { stats[c * 2] = ss[0]; stats[c * 2 + 1] = sq[0]; }
}

__global__ void k_bnpool(const float* __restrict__ x, const float* __restrict__ stats,
                         const float* __restrict__ g, const float* __restrict__ b,
                         float* __restrict__ out, int S) {
  int So = S >> 1;
  int NPo = So * So * So, NP = S * S * S;
  int idx = blockIdx.x * blockDim.x + threadIdx.x;
  if (idx >= 36 * NPo) return;
  int c = idx / NPo, p = idx % NPo;
  int d = (p / (So * So)) * 2, h = ((p / So) % So) * 2, w = (p % So) * 2;
  float inv = 1.f / (float)NP;
  float mean = stats[2 * c] * inv;
  float var = stats[2 * c + 1] * inv - mean * mean;
  float sc = g[c] * rsqrtf(var + 1e-5f);
  float sh = b[c] - mean * sc;
  float m = -INFINITY;
#pragma unroll
  for (int i = 0; i < 2; ++i)
#pragma unroll
    for (int j = 0; j < 2; ++j)
#pragma unroll
      for (int l = 0; l < 2; ++l) {
        float v = x[c * NP + ((d + i) * S + (h + j)) * S + (w + l)];
        m = fmaxf(m, v * sc + sh);
      }
  out[idx] = fmaxf(m, 0.f);
}

// ---------------------------------------------------------------------------
// 27-tap relative-position attention (k=3, pad=1)
// ---------------------------------------------------------------------------
__global__ void k_attn(const float* __restrict__ q, const float* __restrict__ kc,
                       const float* __restrict__ vc, const float* __restrict__ rd,
                       const float* __restrict__ rh, const float* __restrict__ rw,
                       float* __restrict__ out, int S) {
  int NP = S * S * S;
  int idx = blockIdx.x * blockDim.x + threadIdx.x;
  if (idx >= 36 * NP) return;
  int c = idx / NP, p = idx % NP;
  int d = p / (S * S), h = (p / S) % S, w = p % S;
  float qv = q[idx];
  const float* rel; int mode;
  if (c < 12)      { rel = rd + c * 3;        mode = 0; }
  else if (c < 24) { rel = rh + (c - 12) * 3; mode = 1; }
  else             { rel = rw + (c - 24) * 3; mode = 2; }
  float lg[27], vv[27];
  float mx = -INFINITY;
#pragma unroll
  for (int i = 0; i < 3; ++i)
#pragma unroll
    for (int j = 0; j < 3; ++j)
#pragma unroll
      for (int l = 0; l < 3; ++l) {
        int dd = d + i - 1, hh = h + j - 1, ww = w + l - 1;
        float kv = 0.f, vval = 0.f;
        if (dd >= 0 && dd < S && hh >= 0 && hh < S && ww >= 0 && ww < S) {
          int np = (dd * S + hh) * S + ww;
          kv = kc[c * NP + np];
          vval = vc[c * NP + np];
        }
        int ksel = (mode == 0) ? i : (mode == 1 ? j : l);
        kv += rel[ksel];
        int t = (i * 3 + j) * 3 + l;
        float L = qv * kv;
        lg[t] = L; vv[t] = vval;
        mx = fmaxf(mx, L);
      }
  float se = 0.f, num = 0.f;
#pragma unroll
  for (int t = 0; t < 27; ++t) {
    float e = __expf(lg[t] - mx);
    se += e;
    num += e * vv[t];
  }
  out[idx] = num / se;
}

// ---------------------------------------------------------------------------
// Channel attention (one block), channel scale, spatial attention, gate+skip
// ---------------------------------------------------------------------------
__global__ void k_chatt(const float* __restrict__ x, const float* __restrict__ w1,
                        const float* __restrict__ w2, float* __restrict__ s, int NP) {
  __shared__ float mean[36], mxv[36], hm[2], hx[2];
  int t = threadIdx.x;
  if (t < 36) {
    float sm = 0.f, mx = -INFINITY;
    for (int p = 0; p < NP; ++p) {
      float v = x[t * NP + p];
      sm += v; mx = fmaxf(mx, v);
    }
    mean[t] = sm / (float)NP; mxv[t] = mx;
  }
  __syncthreads();
  if (t < 2) {
    float a = 0.f, b = 0.f;
    for (int c = 0; c < 36; ++c) {
      a += w1[t * 36 + c] * mean[c];
      b += w1[t * 36 + c] * mxv[c];
    }
    hm[t] = fmaxf(a, 0.f); hx[t] = fmaxf(b, 0.f);
  }
  __syncthreads();
  if (t < 36) {
    float a = 0.f;
    for (int hh = 0; hh < 2; ++hh) a += w2[t * 2 + hh] * (hm[hh] + hx[hh]);
    s[t] = 1.f / (1.f + __expf(-a));
  }
}

__global__ void k_scale_ch(const float* __restrict__ x, const float* __restrict__ s,
                           float* __restrict__ y, int NP) {
  int idx = blockIdx.x * blockDim.x + threadIdx.x;
  if (idx >= 36 * NP) return;
  y[idx] = x[idx] * s[idx / NP];
}

__global__ void k_featstat(const float* __restrict__ x, float* __restrict__ feat, int NP) {
  int p = blockIdx.x * blockDim.x + threadIdx.x;
  if (p >= NP) return;
  float sm = 0.f, mx = -INFINITY;
  for (int c = 0; c < 36; ++c) {
    float v = x[c * NP + p];
    sm += v; mx = fmaxf(mx, v);
  }
  feat[p] = sm / 36.f;
  feat[NP + p] = mx;
}

__global__ void k_spatconv(const float* __restrict__ feat, const float* __restrict__ w,
                           float* __restrict__ gate, int S) {
  int NP = S * S * S;
  int p = blockIdx.x * blockDim.x + threadIdx.x;
  if (p >= NP) return;
  int d = p / (S * S), h = (p / S) % S, wd = p % S;
  float acc = 0.f;
  for (int ci = 0; ci < 2; ++ci)
    for (int kd = 0; kd < 7; ++kd) {
      int dd = d + kd - 3; if (dd < 0 || dd >= S) continue;
      for (int kh = 0; kh < 7; ++kh) {
        int hh = h + kh - 3; if (hh < 0 || hh >= S) continue;
        for (int kw = 0; kw < 7; ++kw) {
          int ww = wd + kw - 3; if (ww < 0 || ww >= S) continue;
          acc += feat[ci * NP + (dd * S + hh) * S + ww] * w[((ci * 7 + kd) * 7 + kh) * 7 + kw];
        }
      }
    }
  gate[p] = 1.f / (1.f + __expf(-acc));
}

// out[c,p] = y[c,p]*gate[p] + maxpool4(skip)[c,p]; skip spatial = 4*S
__global__ void k_gate_addskip(const float* __restrict__ y, const float* __restrict__ gate,
                               const float* __restrict__ skip, float* __restrict__ out, int S) {
  int NP = S * S * S;
  int idx = blockIdx.x * blockDim.x + threadIdx.x;
  if (idx >= 36 * NP) return;
  int c = idx / NP, p = idx % NP;
  int d = (p / (S * S)) * 4, h = ((p / S) % S) * 4, w = (p % S) * 4;
  int Si = 4 * S, NPi = Si * Si * Si;
  float m = -INFINITY;
#pragma unroll
  for (int i = 0; i < 4; ++i)
#pragma unroll
    for (int j = 0; j < 4; ++j)
#pragma unroll
      for (int l = 0; l < 4; ++l)
        m = fmaxf(m, skip[c * NPi + ((d + i) * Si + (h + j)) * Si + (w + l)]);
  out[idx] = y[idx] * gate[p] + m;
}

// ---------------------------------------------------------------------------
extern "C" void kernel_launch(void* const* d_in, const int* in_sizes, int n_in,
                              void* d_out, int out_size, void* d_ws, size_t ws_size,
                              hipStream_t stream) {
  (void)in_sizes; (void)n_in; (void)out_size; (void)ws_size;
  const float** in = (const float**)d_in;
  const float* data    = in[0];
  const float* s_wq    = in[1];
  const float* s_wk    = in[2];
  const float* s_wv    = in[3];
  const float* emb_a   = in[4];
  const float* emb_b   = in[5];
  const float* emb_c   = in[6];
  const float* emb_mix = in[7];
  // layers 2..5: base 8 + (L-2)*6 -> wq,wk,wv,rel_d,rel_h,rel_w
  // bn: g at 32+(L-1)*2, b at 33+(L-1)*2
  const float* ca1_w1 = in[42]; const float* ca1_w2 = in[43]; const float* sa1_w = in[44];
  const float* ca2_w1 = in[45]; const float* ca2_w2 = in[46]; const float* sa2_w = in[47];
  const float* li1_w = in[48]; const float* li1_b = in[49];
  const float* li2_w = in[50]; const float* li2_b = in[51];
  const float* li3_w = in[52]; const float* li3_b = in[53];
  float* dout = (float*)d_out;

  // workspace layout (floats)
  float* w = (float*)d_ws;
  size_t o = 0;
  float* buf_stem = w + o; o += 36 * 32768;   // stem out 32^3
  float* wveff    = w + o; o += 36 * 64;
  float* stats    = w + o; o += 72;
  float* x16      = w + o; o += 36 * 4096;    // skip1
  float* qkv      = w + o; o += 108 * 4096;
  float* att      = w + o; o += 36 * 4096;
  float* x8       = w + o; o += 36 * 512;
  float* x4       = w + o; o += 36 * 64;
  float* cas      = w + o; o += 36;
  float* xca      = w + o; o += 36 * 64;
  float* feat2    = w + o; o += 2 * 64;
  float* gate     = w + o; o += 64;
  float* x4b      = w + o; o += 36 * 64;      // skip2
  float* x2       = w + o; o += 36 * 8;
  float* x1       = w + o; o += 36;
  float* xfin     = w + o; o += 36;
  float* fc1      = w + o; o += 512;

  auto gemm = [&](const float* A, const float* B, float* C, int M, int N, int K,
                  const float* bias, int relu) {
    int mt = (M + 15) / 16, nt = (N + 15) / 16;
    k_gemm_wmma<<<dim3(mt * nt), dim3(32), 0, stream>>>(A, B, C, M, N, K, bias, relu);
  };
  auto attn_layer = [&](int L, const float* xin, int S, float* xout) {
    int NP = S * S * S;
    int base = 8 + (L - 2) * 6;
    const float* wq = in[base + 0];
    const float* wk = in[base + 1];
    const float* wv = in[base + 2];
    const float* rd = in[base + 3];
    const float* rh = in[base + 4];
    const float* rw = in[base + 5];
    gemm(wq, xin, qkv, 36, NP, 36, nullptr, 0);
    gemm(wk, xin, qkv + 36 * NP, 36, NP, 36, nullptr, 0);
    gemm(wv, xin, qkv + 72 * NP, 36, NP, 36, nullptr, 0);
    int tot = 36 * NP;
    k_attn<<<(tot + 255) / 256, 256, 0, stream>>>(qkv, qkv + 36 * NP, qkv + 72 * NP,
                                                  rd, rh, rw, att, S);
    k_bnstats<<<36, 256, 0, stream>>>(att, stats, NP);
    const float* g = in[32 + (L - 1) * 2];
    const float* b = in[33 + (L - 1) * 2];
    int So = S >> 1;
    int toto = 36 * So * So * So;
    k_bnpool<<<(toto + 255) / 256, 256, 0, stream>>>(att, stats, g, b, xout, S);
  };

  // ---- stem (32^3) + BN1 + pool -> x16 (16^3) ----
  k_wveff<<<1, 64, 0, stream>>>(emb_a, emb_b, emb_c, emb_mix, s_wv, wveff);
  {
    int tot = 36 * 32768;
    k_stem<<<(tot + 255) / 256, 256, 0, stream>>>(data, s_wq, s_wk, wveff, buf_stem);
    k_bnstats<<<36, 256, 0, stream>>>(buf_stem, stats, 32768);
    int toto = 36 * 4096;
    k_bnpool<<<(toto + 255) / 256, 256, 0, stream>>>(buf_stem, stats, in[32], in[33], x16, 32);
  }

  // ---- layers 2,3 ----
  attn_layer(2, x16, 16, x8);
  attn_layer(3, x8, 8, x4);

  // ---- CBAM block 1 at 4^3, skip = x16 pooled 4x ----
  k_chatt<<<1, 64, 0, stream>>>(x4, ca1_w1, ca1_w2, cas, 64);
  k_scale_ch<<<(36 * 64 + 255) / 256, 256, 0, stream>>>(x4, cas, xca, 64);
  k_featstat<<<1, 64, 0, stream>>>(xca, feat2, 64);
  k_spatconv<<<1, 64, 0, stream>>>(feat2, sa1_w, gate, 4);
  k_gate_addskip<<<(36 * 64 + 255) / 256, 256, 0, stream>>>(xca, gate, x16, x4b, 4);

  // ---- layers 4,5 ----
  attn_layer(4, x4b, 4, x2);
  attn_layer(5, x2, 2, x1);

  // ---- CBAM block 2 at 1^3, skip = x4b pooled 4x ----
  k_chatt<<<1, 64, 0, stream>>>(x1, ca2_w1, ca2_w2, cas, 1);
  k_scale_ch<<<1, 64, 0, stream>>>(x1, cas, xca, 1);
  k_featstat<<<1, 32, 0, stream>>>(xca, feat2, 1);
  k_spatconv<<<1, 32, 0, stream>>>(feat2, sa2_w, gate, 1);
  k_gate_addskip<<<1, 64, 0, stream>>>(xca, gate, x4b, xfin, 1);

  // ---- FC head (WMMA GEMV): out -> dout[0:2], feat -> dout[2:258] ----
  gemm(li1_w, xfin, fc1, 512, 1, 36, li1_b, 1);
  gemm(li2_w, fc1, dout + 2, 256, 1, 512, li2_b, 1);
  gemm(li3_w, dout + 2, dout, 2, 1, 256, li3_b, 0);
}